// FilteredNoise_75265006895488
// MI455X (gfx1250) — compile-verified
//
#include <hip/hip_runtime.h>

typedef __attribute__((ext_vector_type(2))) float v2f;
typedef __attribute__((ext_vector_type(8))) float v8f;

#define WIN        3072
#define HOP        192
#define HALF       1536
#define T_FRAMES   1000
#define NBINS      1536                 // kidx 0..1535 -> frequency k = kidx+1
#define NBANDS     128
#define NOISE_LEN  ((T_FRAMES - 1) * HOP)   // 191808
#define PAD_LEN    (NOISE_LEN + WIN)        // 194880
#define TWO_PI     6.28318530717958647692f

// ---- workspace layout (float offsets) ----
#define OFF_COSN  ((size_t)0)                         // [WIN][NBINS]  B for analysis (n-major)
#define OFF_SINN  (OFF_COSN + (size_t)WIN * NBINS)
#define OFF_COSK  (OFF_SINN + (size_t)WIN * NBINS)    // [NBINS][WIN]  B for synthesis (k-major)
#define OFF_SINK  (OFF_COSK + (size_t)NBINS * WIN)
#define OFF_HANN  (OFF_SINK + (size_t)NBINS * WIN)    // [WIN]
#define OFF_PAD   (OFF_HANN + (size_t)WIN)            // [PAD_LEN]   (per-batch, reused)
#define OFF_ZRE   (OFF_PAD  + (size_t)PAD_LEN)        // [T_FRAMES][NBINS]
#define OFF_ZIM   (OFF_ZRE  + (size_t)T_FRAMES * NBINS)
#define OFF_Y     (OFF_ZIM  + (size_t)T_FRAMES * NBINS) // [T_FRAMES][WIN]

#define MTILES_PER_WAVE 4               // register blocking: 4 t-tiles share one B fragment

__device__ __forceinline__ v8f wmma_f32(v2f a, v2f b, v8f c) {
  // V_WMMA_F32_16X16X4_F32 : D = A(16x4) x B(4x16) + C, fp32 matrix pipe
  return __builtin_amdgcn_wmma_f32_16x16x4_f32(false, a, false, b, (short)0, c,
                                               false, false);
}

// ---- table generation: cos/sin in both orientations + hann ----
__global__ void gen_tables_kernel(float* ws) {
  size_t idx = (size_t)blockIdx.x * blockDim.x + threadIdx.x;
  size_t total = (size_t)NBINS * WIN;
  if (idx < total) {
    int kidx = (int)(idx / WIN);
    int n    = (int)(idx % WIN);
    int k    = kidx + 1;
    int m    = (k * n) % WIN;                 // exact integer angle reduction
    float ang = TWO_PI * (float)m / (float)WIN;
    float c = cosf(ang);
    float s = sinf(ang);
    ws[OFF_COSK + (size_t)kidx * WIN + n] = c;
    ws[OFF_SINK + (size_t)kidx * WIN + n] = s;
    ws[OFF_COSN + (size_t)n * NBINS + kidx] = c;
    ws[OFF_SINN + (size_t)n * NBINS + kidx] = s;
  }
  if (idx < WIN) {
    ws[OFF_HANN + idx] =
        0.5f - 0.5f * cosf(TWO_PI * (float)idx / (float)(WIN - 1));
  }
}

// ---- zero-pad one batch of noise: padded[i] = noise[i-HALF] or 0 ----
__global__ void pad_kernel(const float* __restrict__ noise_b, float* __restrict__ padded) {
  int i = blockIdx.x * blockDim.x + threadIdx.x;
  if (i >= PAD_LEN) return;
  float v = 0.0f;
  int src = i - HALF;
  if (src >= 0 && src < NOISE_LEN) v = noise_b[src];
  padded[i] = v;
}

// ---- stage 1: Zre/Zim[t,k] = sum_n frame[t,n] * cos/sin(2*pi*k*n/N) ----
// A = frames (row t at padded + t*HOP, lda=HOP, overlapping rows), K = WIN
// B = cosN/sinN (n-major, ldb = NBINS).
// Each wave: one 16-wide k-tile x 4 consecutive 16-row t-tiles; B fragments
// (cos+sin) are loaded once per K-step and reused by 8 WMMAs.
__global__ void analysis_gemm_kernel(const float* __restrict__ padded,
                                     const float* __restrict__ cosN,
                                     const float* __restrict__ sinN,
                                     float* __restrict__ Zre,
                                     float* __restrict__ Zim) {
  int lane = threadIdx.x & 31;
  int wv   = threadIdx.x >> 5;        // 4 waves per block, one k-tile each
  int hs   = lane >> 4;               // half-wave select -> K pair {0,1} vs {2,3}
  int lr   = lane & 15;
  int col0 = (blockIdx.x * 4 + wv) * 16;         // k tile
  int mt0  = blockIdx.y * MTILES_PER_WAVE;       // first t tile of this wave

  const float* arow[MTILES_PER_WAVE];
#pragma unroll
  for (int i = 0; i < MTILES_PER_WAVE; ++i) {
    int t = (mt0 + i) * 16 + lr;
    if (t > T_FRAMES - 1) t = T_FRAMES - 1;
    arow[i] = padded + (size_t)t * HOP;
  }

  v8f accC[MTILES_PER_WAVE] = {};
  v8f accS[MTILES_PER_WAVE] = {};
#pragma unroll 2
  for (int k = 0; k < WIN; k += 4) {
    const float* bc = cosN + (size_t)(k + 2 * hs) * NBINS + col0 + lr;
    const float* bs = sinN + (size_t)(k + 2 * hs) * NBINS + col0 + lr;
    v2f bC; bC.x = bc[0]; bC.y = bc[NBINS];
    v2f bS; bS.x = bs[0]; bS.y = bs[NBINS];
#pragma unroll
    for (int i = 0; i < MTILES_PER_WAVE; ++i) {
      v2f a = *(const v2f*)(arow[i] + k + 2 * hs);      // A[t][k+2hs .. +1]
      accC[i] = wmma_f32(a, bC, accC[i]);
      accS[i] = wmma_f32(a, bS, accS[i]);
    }
  }
#pragma unroll
  for (int i = 0; i < MTILES_PER_WAVE; ++i) {
#pragma unroll
    for (int r = 0; r < 8; ++r) {
      int tm = (mt0 + i) * 16 + r + 8 * hs;             // D row mapping
      if (tm < T_FRAMES) {
        Zre[(size_t)tm * NBINS + col0 + lr] = accC[i][r];
        Zim[(size_t)tm * NBINS + col0 + lr] = accS[i][r];
      }
    }
  }
}

// ---- stage 2: apply per-(t, band) filter gain and rfft normalization ----
__global__ void scale_kernel(const float* __restrict__ fb_b,  // [NBANDS][T_FRAMES]
                             float* __restrict__ Zre, float* __restrict__ Zim) {
  size_t idx = (size_t)blockIdx.x * blockDim.x + threadIdx.x;
  if (idx >= (size_t)T_FRAMES * NBINS) return;
  int t    = (int)(idx / NBINS);
  int kidx = (int)(idx % NBINS);
  int band = kidx / 12;                                   // (k-1)//12
  float ak = (kidx == NBINS - 1 ? 1.0f : 2.0f) / (float)WIN;
  float g  = ak * fb_b[(size_t)band * T_FRAMES + t];
  Zre[idx] *= g;
  Zim[idx] *= g;
}

// ---- stage 3: Y[t,n] = sum_k Zre'[t,k]*cosK[k,n] + Zim'[t,k]*sinK[k,n] ----
// Same register blocking: 4 t-tiles per wave reuse the B fragments.
__global__ void synthesis_gemm_kernel(const float* __restrict__ Zre,
                                      const float* __restrict__ Zim,
                                      const float* __restrict__ cosK,
                                      const float* __restrict__ sinK,
                                      float* __restrict__ Y) {
  int lane = threadIdx.x & 31;
  int wv   = threadIdx.x >> 5;
  int hs   = lane >> 4;
  int lr   = lane & 15;
  int col0 = (blockIdx.x * 4 + wv) * 16;         // n tile
  int mt0  = blockIdx.y * MTILES_PER_WAVE;       // first t tile

  const float* are[MTILES_PER_WAVE];
  const float* aim[MTILES_PER_WAVE];
#pragma unroll
  for (int i = 0; i < MTILES_PER_WAVE; ++i) {
    int t = (mt0 + i) * 16 + lr;
    if (t > T_FRAMES - 1) t = T_FRAMES - 1;
    are[i] = Zre + (size_t)t * NBINS;
    aim[i] = Zim + (size_t)t * NBINS;
  }

  v8f acc[MTILES_PER_WAVE] = {};
#pragma unroll 2
  for (int k = 0; k < NBINS; k += 4) {
    const float* bc = cosK + (size_t)(k + 2 * hs) * WIN + col0 + lr;
    const float* bs = sinK + (size_t)(k + 2 * hs) * WIN + col0 + lr;
    v2f bC; bC.x = bc[0]; bC.y = bc[WIN];
    v2f bS; bS.x = bs[0]; bS.y = bs[WIN];
#pragma unroll
    for (int i = 0; i < MTILES_PER_WAVE; ++i) {
      v2f aR = *(const v2f*)(are[i] + k + 2 * hs);
      v2f aI = *(const v2f*)(aim[i] + k + 2 * hs);
      acc[i] = wmma_f32(aR, bC, acc[i]);
      acc[i] = wmma_f32(aI, bS, acc[i]);
    }
  }
#pragma unroll
  for (int i = 0; i < MTILES_PER_WAVE; ++i) {
#pragma unroll
    for (int r = 0; r < 8; ++r) {
      int tm = (mt0 + i) * 16 + r + 8 * hs;
      if (tm < T_FRAMES)
        Y[(size_t)tm * WIN + col0 + lr] = acc[i][r];
    }
  }
}

// ---- stage 4: window + overlap-add as a deterministic gather ----
__global__ void ola_kernel(const float* __restrict__ Y,
                           const float* __restrict__ hann,
                           float* __restrict__ out_b) {
  int l = blockIdx.x * blockDim.x + threadIdx.x;
  if (l >= NOISE_LEN) return;
  int p  = l + HALF;           // position in padded output
  int m  = p / HOP;            // m in [8, 1007]
  int r  = p % HOP;
  int qlo = m - (T_FRAMES - 1); if (qlo < 0) qlo = 0;
  int qhi = m < 15 ? m : 15;
  float s = 0.0f;
  for (int q = qlo; q <= qhi; ++q) {
    int j = r + HOP * q;                         // sample within frame (m-q)
    s += hann[j] * Y[(size_t)(m - q) * WIN + j];
  }
  out_b[l] = s;
}

extern "C" void kernel_launch(void* const* d_in, const int* in_sizes, int n_in,
                              void* d_out, int out_size, void* d_ws, size_t ws_size,
                              hipStream_t stream) {
  (void)in_sizes; (void)n_in; (void)out_size; (void)ws_size;
  const float* fb    = (const float*)d_in[0];   // (8,1,128,1000)
  const float* noise = (const float*)d_in[1];   // (8,1,191808)
  float* out = (float*)d_out;                   // (8,1,191808)
  float* ws  = (float*)d_ws;

  float* cosN = ws + OFF_COSN;
  float* sinN = ws + OFF_SINN;
  float* cosK = ws + OFF_COSK;
  float* sinK = ws + OFF_SINK;
  float* hann = ws + OFF_HANN;
  float* pad  = ws + OFF_PAD;
  float* Zre  = ws + OFF_ZRE;
  float* Zim  = ws + OFF_ZIM;
  float* Y    = ws + OFF_Y;

  {
    size_t total = (size_t)NBINS * WIN;
    int blocks = (int)((total + 255) / 256);
    gen_tables_kernel<<<blocks, 256, 0, stream>>>(ws);
  }

  int tilesM  = (T_FRAMES + 15) / 16;                       // 63
  int groupsM = (tilesM + MTILES_PER_WAVE - 1) / MTILES_PER_WAVE;  // 16
  dim3 g1(NBINS / 64, groupsM);   // 24 x 16, 4 waves/block, 4 t-tiles/wave
  dim3 g3(WIN / 64,   groupsM);   // 48 x 16
  for (int b = 0; b < 8; ++b) {
    const float* noise_b = noise + (size_t)b * NOISE_LEN;
    const float* fb_b    = fb + (size_t)b * NBANDS * T_FRAMES;
    float* out_b         = out + (size_t)b * NOISE_LEN;

    pad_kernel<<<(PAD_LEN + 255) / 256, 256, 0, stream>>>(noise_b, pad);
    analysis_gemm_kernel<<<g1, 128, 0, stream>>>(pad, cosN, sinN, Zre, Zim);
    {
      size_t tot = (size_t)T_FRAMES * NBINS;
      scale_kernel<<<(int)((tot + 255) / 256), 256, 0, stream>>>(fb_b, Zre, Zim);
    }
    synthesis_gemm_kernel<<<g3, 128, 0, stream>>>(Zre, Zim, cosK, sinK, Y);
    ola_kernel<<<(NOISE_LEN + 255) / 256, 256, 0, stream>>>(Y, hann, out_b);
  }
}